// MyParticleNetwork_58841051955745
// MI455X (gfx1250) — compile-verified
//
#include <hip/hip_runtime.h>
#include <math.h>

#define N_FLUID 6000
#define N_BOX   3000
#define K_MAX   80
#define KS      4
#define NCELL   64
#define RADIUS  0.1125f
#define MAXC    256
#define CHUNK   2000   // particles per scatter/GEMM chunk (S chunk = 2000*64*96*4B = 49.2MB)

typedef __attribute__((ext_vector_type(2))) float v2f;
typedef __attribute__((ext_vector_type(8))) float v8f;

__device__ __forceinline__ float signf(float a) {
    return (a > 0.f) ? 1.f : ((a < 0.f) ? -1.f : 0.f);
}

// Volume-preserving ball->cube mapping (open3d ContinuousConv convention).
__device__ void ball_to_cube(float x, float y, float z, float& u, float& v, float& w) {
    const float eps = 1e-12f;
    float sq   = x*x + y*y + z*z;
    float nrm  = sqrtf(sq);
    float rxy2 = x*x + y*y;
    bool  polar = (1.25f * z * z) > rxy2;
    float s_pol = sqrtf(3.0f * nrm / (nrm + fabsf(z) + eps));
    float s_eq  = nrm / sqrtf(rxy2 + eps);
    float xc = polar ? s_pol * x : s_eq * x;
    float yc = polar ? s_pol * y : s_eq * y;
    float zc = polar ? signf(z) * nrm : 1.5f * z * s_eq;
    if (sq < 1e-12f) { xc = 0.f; yc = 0.f; zc = 0.f; }
    // cylinder -> cube (disc -> square on xy)
    float rxy = sqrtf(xc*xc + yc*yc);
    const float FP = 4.0f / 3.14159265358979323846f;
    float safe_x = (fabsf(xc) < eps) ? eps : xc;
    float safe_y = (fabsf(yc) < eps) ? eps : yc;
    bool  cond = fabsf(xc) >= fabsf(yc);
    float u1 = signf(xc) * rxy;
    float v1 = u1 * FP * atanf(yc / safe_x);
    float v2_ = signf(yc) * rxy;
    float u2 = v2_ * FP * atanf(xc / safe_y);
    u = cond ? u1 : u2;
    v = cond ? v1 : v2_;
    if (rxy < eps) { u = 0.f; v = 0.f; }
    w = zc;
}

// ---------------------------------------------------------------------------
// Neighbor search: one block per query particle. Collect all in-radius
// candidates into LDS, then rank-select the K_MAX closest (slot = rank on d^2,
// deterministic regardless of LDS atomic fill order) and emit per-neighbor
// geometry: index, trilinear cell base, t-fractions and poly6 window weight.
// ---------------------------------------------------------------------------
__global__ void neighbor_kernel(const float* __restrict__ qpos,
                                const float* __restrict__ cpos, int nc,
                                int ignore_self,
                                int* __restrict__ n_idx, int* __restrict__ n_base,
                                float* __restrict__ n_t, float* __restrict__ n_win) {
    int q = blockIdx.x;
    __shared__ int   s_cnt;
    __shared__ int   s_idx[MAXC];
    __shared__ float s_d2[MAXC];
    float qx = qpos[q*3+0], qy = qpos[q*3+1], qz = qpos[q*3+2];
    if (threadIdx.x == 0) s_cnt = 0;
    __syncthreads();
    const float R2 = RADIUS * RADIUS;
    for (int i = threadIdx.x; i < nc; i += blockDim.x) {
        if (ignore_self && i == q) continue;
        float dx = cpos[i*3+0] - qx;
        float dy = cpos[i*3+1] - qy;
        float dz = cpos[i*3+2] - qz;
        float d2 = dx*dx + dy*dy + dz*dz;
        if (d2 <= R2) {
            int slot = atomicAdd(&s_cnt, 1);
            if (slot < MAXC) { s_idx[slot] = i; s_d2[slot] = d2; }
        }
    }
    __syncthreads();
    int cnt  = min(s_cnt, MAXC);
    int kept = min(cnt, K_MAX);
    size_t ob = (size_t)q * K_MAX;
    for (int i = threadIdx.x; i < cnt; i += blockDim.x) {
        float di = s_d2[i];
        int rank = 0;
        for (int jj = 0; jj < cnt; ++jj) {
            float dj = s_d2[jj];
            if (dj < di || (dj == di && jj < i)) rank++;
        }
        if (rank >= K_MAX) continue;
        int j = s_idx[i];
        float rx = (cpos[j*3+0] - qx) * (1.0f / RADIUS);
        float ry = (cpos[j*3+1] - qy) * (1.0f / RADIUS);
        float rz = (cpos[j*3+2] - qz) * (1.0f / RADIUS);
        float r2 = rx*rx + ry*ry + rz*rz;
        float w1 = 1.0f - r2;
        float win = fminf(fmaxf(w1*w1*w1, 0.0f), 1.0f);
        float u, v, wc;
        ball_to_cube(rx, ry, rz, u, v, wc);
        float gx = fminf(fmaxf((u  + 1.f) * 0.5f * (KS - 1), 0.f), (float)(KS - 1));
        float gy = fminf(fmaxf((v  + 1.f) * 0.5f * (KS - 1), 0.f), (float)(KS - 1));
        float gz = fminf(fmaxf((wc + 1.f) * 0.5f * (KS - 1), 0.f), (float)(KS - 1));
        int ix = min(max((int)floorf(gx), 0), KS - 2);
        int iy = min(max((int)floorf(gy), 0), KS - 2);
        int iz = min(max((int)floorf(gz), 0), KS - 2);
        n_idx[ob + rank]  = j;
        n_base[ob + rank] = iz * (KS*KS) + iy * KS + ix;   // cell = ((z)*4+(y))*4+(x)
        n_win[ob + rank]  = win;
        n_t[(ob + rank)*3 + 0] = gx - (float)ix;
        n_t[(ob + rank)*3 + 1] = gy - (float)iy;
        n_t[(ob + rank)*3 + 2] = gz - (float)iz;
    }
    for (int k = kept + threadIdx.x; k < K_MAX; k += blockDim.x) {
        n_idx[ob + k] = 0; n_base[ob + k] = 0; n_win[ob + k] = 0.f;
        n_t[(ob + k)*3 + 0] = 0.f; n_t[(ob + k)*3 + 1] = 0.f; n_t[(ob + k)*3 + 2] = 0.f;
    }
}

// ---------------------------------------------------------------------------
// Scatter: one block per query particle, builds S[n] = [64 cells, F] in LDS.
// Race-free: thread f owns feature column f; neighbors walked sequentially.
// ---------------------------------------------------------------------------
__global__ void scatter_kernel(const float* __restrict__ feats, int F,
                               const int* __restrict__ nidx,
                               const int* __restrict__ nbase,
                               const float* __restrict__ nt,
                               const float* __restrict__ nwin,
                               int m0, float* __restrict__ S) {
    int n = m0 + blockIdx.x;
    __shared__ float Sloc[NCELL * 96];      // max F = 96 -> 24.6 KB
    int tot = NCELL * F;
    for (int i = threadIdx.x; i < tot; i += blockDim.x) Sloc[i] = 0.f;
    __syncthreads();
    int f = threadIdx.x;
    if (f < F) {
        size_t nb = (size_t)n * K_MAX;
        for (int k = 0; k < K_MAX; ++k) {
            float w = nwin[nb + k];
            if (w == 0.0f) continue;
            int   j    = nidx[nb + k];
            int   base = nbase[nb + k];
            float tx = nt[(nb + k)*3 + 0];
            float ty = nt[(nb + k)*3 + 1];
            float tz = nt[(nb + k)*3 + 2];
            float wv = w * feats[(size_t)j * F + f];
            float wx0 = (1.f - tx) * wv, wx1 = tx * wv;
            float wy0 = 1.f - ty, wy1 = ty;
            float wz0 = 1.f - tz, wz1 = tz;
            // corner offset = dz*16 + dy*4 + dx
            Sloc[(base +  0)*F + f] += wx0 * wy0 * wz0;
            Sloc[(base +  1)*F + f] += wx1 * wy0 * wz0;
            Sloc[(base +  4)*F + f] += wx0 * wy1 * wz0;
            Sloc[(base +  5)*F + f] += wx1 * wy1 * wz0;
            Sloc[(base + 16)*F + f] += wx0 * wy0 * wz1;
            Sloc[(base + 17)*F + f] += wx1 * wy0 * wz1;
            Sloc[(base + 20)*F + f] += wx0 * wy1 * wz1;
            Sloc[(base + 21)*F + f] += wx1 * wy1 * wz1;
        }
    }
    __syncthreads();
    float* Sg = S + (size_t)blockIdx.x * tot;
    for (int i = threadIdx.x; i < tot; i += blockDim.x) Sg[i] = Sloc[i];
}

// ---------------------------------------------------------------------------
// Branch-free fp32 WMMA GEMM: one wave32 per 16x16 output tile.
// REQUIRES: M % 16 == 0, N % 16 == 0, K % 4 == 0 (guaranteed by caller; the
// O=3 output layer is zero-padded to 16 columns).
// C[m,n] (+)= sum_k A[m*lda+k] * B[k*ldb+n] + bias[n]
// A frag: lane<16 -> M=lane, K={0,1}; lane>=16 -> K={2,3}  (one b64 load).
// B frag: lane<16 -> N=lane, K={0,1}; lane>=16 -> K={2,3}  (two b32 loads).
// D frag: lane%16 -> N; vgpr r -> M = r + 8*(lane>=16).
// ---------------------------------------------------------------------------
__global__ void __launch_bounds__(32)
gemm_wmma(const float* __restrict__ A, int lda,
          const float* __restrict__ B, int ldb,
          const float* __restrict__ bias,
          float* __restrict__ C, int ldc,
          int K, int accumulate) {
    int lane = threadIdx.x;
    int half = lane >> 4;          // 0 or 1
    int l16  = lane & 15;
    const float* Arow = A + (size_t)(blockIdx.x * 16 + l16) * lda + 2 * half;
    const float* Bcol = B + (size_t)(2 * half) * ldb + (blockIdx.y * 16 + l16);
    const size_t bstep = (size_t)4 * ldb;
    v8f acc = {};
#pragma unroll 4
    for (int k = 0; k < K; k += 4) {
        v2f a = *(const v2f*)Arow;           // 8B-aligned: lda%4==0, offset even
        v2f b;
        b.x = Bcol[0];
        b.y = Bcol[ldb];
        acc = __builtin_amdgcn_wmma_f32_16x16x4_f32(
            /*neg_a=*/false, a, /*neg_b=*/false, b,
            /*c_mod=*/(short)0, acc, /*reuse_a=*/false, /*reuse_b=*/false);
        Arow += 4;
        Bcol += bstep;
    }
    int an = blockIdx.y * 16 + l16;
    float bv = (bias != nullptr) ? bias[an] : 0.f;
    float* Crow = C + (size_t)(blockIdx.x * 16 + half * 8) * ldc + an;
#pragma unroll
    for (int r = 0; r < 8; ++r) {
        float vv = acc[r] + bv;
        if (accumulate) vv += Crow[(size_t)r * ldc];
        Crow[(size_t)r * ldc] = vv;
    }
}

// --------------------------- elementwise helpers ---------------------------
__global__ void build_ff_kernel(const float* __restrict__ vel, float* __restrict__ ff) {
    int i = blockIdx.x * blockDim.x + threadIdx.x;
    if (i < N_FLUID) {
        ff[i*4 + 0] = 1.0f;
        ff[i*4 + 1] = vel[i*3 + 0];
        ff[i*4 + 2] = vel[i*3 + 1];
        ff[i*4 + 3] = vel[i*3 + 2];
    }
}
__global__ void relu_kernel(const float* __restrict__ x, float* __restrict__ h, int n) {
    int i = blockIdx.x * blockDim.x + threadIdx.x;
    if (i < n) h[i] = fmaxf(x[i], 0.f);
}
__global__ void add_kernel(float* __restrict__ x, const float* __restrict__ y, int n) {
    int i = blockIdx.x * blockDim.x + threadIdx.x;
    if (i < n) x[i] += y[i];
}
// Zero-pad an [K x O] row-major matrix to [K x Opad].
__global__ void pad_cols_kernel(const float* __restrict__ src, int K, int O,
                                float* __restrict__ dst, int Opad) {
    int i = blockIdx.x * blockDim.x + threadIdx.x;
    if (i < K * Opad) {
        int r = i / Opad, c = i % Opad;
        dst[i] = (c < O) ? src[r * O + c] : 0.f;
    }
}
// Extract first 3 of 16 padded columns, apply final 1/128 scale.
__global__ void finalize_kernel(const float* __restrict__ y16, float* __restrict__ out) {
    int i = blockIdx.x * blockDim.x + threadIdx.x;
    if (i < N_FLUID * 3) {
        int n = i / 3, c = i % 3;
        out[i] = y16[n * 16 + c] * (1.0f / 128.0f);
    }
}

// ---------------------------------------------------------------------------
extern "C" void kernel_launch(void* const* d_in, const int* in_sizes, int n_in,
                              void* d_out, int out_size, void* d_ws, size_t ws_size,
                              hipStream_t stream) {
    (void)in_sizes; (void)n_in; (void)out_size; (void)ws_size;
    const float* pos       = (const float*)d_in[0];
    const float* vel       = (const float*)d_in[1];
    const float* box       = (const float*)d_in[2];
    const float* box_feats = (const float*)d_in[3];
    const float* cf_w = (const float*)d_in[4];  const float* cf_b = (const float*)d_in[5];
    const float* co_w = (const float*)d_in[6];  const float* co_b = (const float*)d_in[7];
    const float* d0_w = (const float*)d_in[8];  const float* d0_b = (const float*)d_in[9];
    const float* c1_w = (const float*)d_in[10]; const float* c1_b = (const float*)d_in[11];
    const float* d1_w = (const float*)d_in[12]; const float* d1_b = (const float*)d_in[13];
    const float* c2_w = (const float*)d_in[14]; const float* c2_b = (const float*)d_in[15];
    const float* d2_w = (const float*)d_in[16]; const float* d2_b = (const float*)d_in[17];
    const float* c3_w = (const float*)d_in[18]; const float* c3_b = (const float*)d_in[19];
    const float* d3_w = (const float*)d_in[20]; const float* d3_b = (const float*)d_in[21];

    // ---- workspace carve-out (~78 MB total) ----
    char* wsp = (char*)d_ws;
    auto carve = [&](size_t bytes) {
        void* p = (void*)wsp;
        wsp += (bytes + 255) & ~(size_t)255;
        return p;
    };
    const size_t NPK = (size_t)N_FLUID * K_MAX;
    int*   fn_idx  = (int*)  carve(NPK * sizeof(int));
    int*   fn_base = (int*)  carve(NPK * sizeof(int));
    float* fn_t    = (float*)carve(NPK * 3 * sizeof(float));
    float* fn_win  = (float*)carve(NPK * sizeof(float));
    int*   bn_idx  = (int*)  carve(NPK * sizeof(int));
    int*   bn_base = (int*)  carve(NPK * sizeof(int));
    float* bn_t    = (float*)carve(NPK * 3 * sizeof(float));
    float* bn_win  = (float*)carve(NPK * sizeof(float));
    float* ff      = (float*)carve((size_t)N_FLUID * 4 * sizeof(float));
    float* x96     = (float*)carve((size_t)N_FLUID * 96 * sizeof(float));
    float* h96     = (float*)carve((size_t)N_FLUID * 96 * sizeof(float));
    float* bufA    = (float*)carve((size_t)N_FLUID * 64 * sizeof(float));
    float* bufB    = (float*)carve((size_t)N_FLUID * 64 * sizeof(float));
    float* y16     = (float*)carve((size_t)N_FLUID * 16 * sizeof(float));
    float* c3w_pad = (float*)carve((size_t)NCELL * 64 * 16 * sizeof(float)); // 4096x16
    float* d3w_pad = (float*)carve((size_t)64 * 16 * sizeof(float));
    float* c3b_pad = (float*)carve((size_t)16 * sizeof(float));
    float* d3b_pad = (float*)carve((size_t)16 * sizeof(float));
    float* Sbuf    = (float*)carve((size_t)CHUNK * NCELL * 96 * sizeof(float));

    auto run_gemm = [&](const float* A, int lda, const float* B, int ldb,
                        const float* bias, float* C, int ldc,
                        int M, int K, int O, int accumulate) {
        dim3 grid(M / 16, O / 16);
        gemm_wmma<<<grid, 32, 0, stream>>>(A, lda, B, ldb, bias, C, ldc, K, accumulate);
    };
    auto run_conv = [&](const float* feats, int F,
                        const int* nidx, const int* nbase, const float* nt, const float* nwin,
                        const float* W, const float* bias, float* C, int ldc, int O) {
        int K = NCELL * F;
        for (int m0 = 0; m0 < N_FLUID; m0 += CHUNK) {
            int mc = min(CHUNK, N_FLUID - m0);
            scatter_kernel<<<mc, 128, 0, stream>>>(feats, F, nidx, nbase, nt, nwin, m0, Sbuf);
            run_gemm(Sbuf, K, W, O, bias, C + (size_t)m0 * ldc, ldc, mc, K, O, /*acc=*/0);
        }
    };
    auto ew = [&](int n) { return dim3((n + 255) / 256); };

    // ---- neighbor lists ----
    neighbor_kernel<<<N_FLUID, 128, 0, stream>>>(pos, pos, N_FLUID, 1,
                                                 fn_idx, fn_base, fn_t, fn_win);
    neighbor_kernel<<<N_FLUID, 128, 0, stream>>>(pos, box, N_BOX, 0,
                                                 bn_idx, bn_base, bn_t, bn_win);
    build_ff_kernel<<<ew(N_FLUID), 256, 0, stream>>>(vel, ff);
    // pad the O=3 weights/biases of block 3 to 16 columns
    pad_cols_kernel<<<ew(NCELL*64*16), 256, 0, stream>>>(c3_w, NCELL*64, 3, c3w_pad, 16);
    pad_cols_kernel<<<ew(64*16),       256, 0, stream>>>(d3_w, 64,       3, d3w_pad, 16);
    pad_cols_kernel<<<1, 16, 0, stream>>>(c3_b, 1, 3, c3b_pad, 16);
    pad_cols_kernel<<<1, 16, 0, stream>>>(d3_b, 1, 3, d3b_pad, 16);

    // ---- layer 0: x96 = concat[a_co | a_cf | a_d0] ----
    run_conv(box_feats, 3, bn_idx, bn_base, bn_t, bn_win, co_w, co_b, x96 + 0,  96, 32);
    run_conv(ff,        4, fn_idx, fn_base, fn_t, fn_win, cf_w, cf_b, x96 + 32, 96, 32);
    run_gemm(ff, 4, d0_w, 32, d0_b, x96 + 64, 96, N_FLUID, 4, 32, /*acc=*/0);

    // ---- block 1: x -> bufA (96 -> 64, no residual) ----
    relu_kernel<<<ew(N_FLUID*96), 256, 0, stream>>>(x96, h96, N_FLUID * 96);
    run_conv(h96, 96, fn_idx, fn_base, fn_t, fn_win, c1_w, c1_b, bufA, 64, 64);
    run_gemm(h96, 96, d1_w, 64, d1_b, bufA, 64, N_FLUID, 96, 64, /*acc=*/1);

    // ---- block 2: bufA -> bufB (64 -> 64, residual) ----
    relu_kernel<<<ew(N_FLUID*64), 256, 0, stream>>>(bufA, h96, N_FLUID * 64);
    run_conv(h96, 64, fn_idx, fn_base, fn_t, fn_win, c2_w, c2_b, bufB, 64, 64);
    run_gemm(h96, 64, d2_w, 64, d2_b, bufB, 64, N_FLUID, 64, 64, /*acc=*/1);
    add_kernel<<<ew(N_FLUID*64), 256, 0, stream>>>(bufB, bufA, N_FLUID * 64);

    // ---- block 3: bufB -> y16 (64 -> 16 padded), then extract 3 cols * 1/128 ----
    relu_kernel<<<ew(N_FLUID*64), 256, 0, stream>>>(bufB, h96, N_FLUID * 64);
    run_conv(h96, 64, fn_idx, fn_base, fn_t, fn_win, c3w_pad, c3b_pad, y16, 16, 16);
    run_gemm(h96, 64, d3w_pad, 16, d3b_pad, y16, 16, N_FLUID, 64, 16, /*acc=*/1);
    finalize_kernel<<<ew(N_FLUID*3), 256, 0, stream>>>(y16, (float*)d_out);
}